// EmbedMeanField_6107443495393
// MI455X (gfx1250) — compile-verified
//
#include <hip/hip_runtime.h>

#define LATENT 64
#define FN_ 64
#define FE_ 16
#define LV_ 3

typedef __attribute__((ext_vector_type(2))) float v2f;
typedef __attribute__((ext_vector_type(8))) float v8f;

// ---------------------------------------------------------------------------
// Elementwise helpers (float4 vectorized)
// ---------------------------------------------------------------------------
__global__ void zero_kernel(float* __restrict__ p, long n4) {
  long i = (long)blockIdx.x * blockDim.x + threadIdx.x;
  if (i < n4) ((float4*)p)[i] = make_float4(0.f, 0.f, 0.f, 0.f);
}

__global__ void relu_kernel(const float* __restrict__ in, float* __restrict__ out, long n4) {
  long i = (long)blockIdx.x * blockDim.x + threadIdx.x;
  if (i < n4) {
    float4 v = ((const float4*)in)[i];
    v.x = fmaxf(v.x, 0.f); v.y = fmaxf(v.y, 0.f);
    v.z = fmaxf(v.z, 0.f); v.w = fmaxf(v.w, 0.f);
    ((float4*)out)[i] = v;
  }
}

// ---------------------------------------------------------------------------
// fp32 WMMA tile GEMM: accumulates a 16x16 f32 tile, K steps of 4.
// A-layout (16x4 f32, ISA 7.12.2): lane L holds row M=L%16; VGPR0 = K=(0|2),
// VGPR1 = K=(1|3), second K pair for lanes 16-31.
// B-layout (4x16): row striped across lanes within a VGPR, same half-split.
// ---------------------------------------------------------------------------
template <int K>
__device__ __forceinline__ v8f wmma_f32_gemm(const float* __restrict__ A, int lda,
                                             const float* __restrict__ B, /*ldb=LATENT*/
                                             v8f acc, int lane) {
  const int lrow = lane & 15;
  const int koff = (lane >= 16) ? 2 : 0;
#pragma unroll
  for (int k = 0; k < K; k += 4) {
    v2f a, b;
    a.x = A[(long)lrow * lda + k + koff];
    a.y = A[(long)lrow * lda + k + koff + 1];
    b.x = B[(long)(k + koff) * LATENT + lrow];
    b.y = B[(long)(k + koff + 1) * LATENT + lrow];
    acc = __builtin_amdgcn_wmma_f32_16x16x4_f32(false, a, false, b, (short)0, acc,
                                                false, false);
  }
  return acc;
}

// ---------------------------------------------------------------------------
// msg = node_feat @ W_n2l + b_n2l      ([N,64] x [64,64])
// ---------------------------------------------------------------------------
__global__ __launch_bounds__(256) void node_linear_wmma(
    const float* __restrict__ nf, const float* __restrict__ W,
    const float* __restrict__ bias, float* __restrict__ msg, int N) {
  int wid  = (int)(((long)blockIdx.x * blockDim.x + threadIdx.x) >> 5);
  int lane = threadIdx.x & 31;
  int rowTiles = (N + 15) >> 4;
  if (wid >= rowTiles * (LATENT / 16)) return;   // wave-uniform
  int rt = wid >> 2, ct = wid & 3;
  int row0 = rt << 4, col0 = ct << 4;
  int lrow = lane & 15;

  v8f acc;
  float bval = bias[col0 + lrow];
#pragma unroll
  for (int r = 0; r < 8; ++r) acc[r] = bval;

  int arow = min(row0 + lrow, N - 1);            // clamp for partial tiles
  acc = wmma_f32_gemm<FN_>(nf + ((long)arow - lrow) * FN_, FN_, W + col0, acc, lane);

  int mh = (lane >= 16) ? 8 : 0;
#pragma unroll
  for (int r = 0; r < 8; ++r) {
    int m = row0 + r + mh;
    if (m < N) msg[(long)m * LATENT + col0 + lrow] = acc[r];
  }
}

// ---------------------------------------------------------------------------
// edge_lin = edge_feat @ W_e2l + b_e2l ; atomic scatter-add into msg[edge_dst]
// ---------------------------------------------------------------------------
__global__ __launch_bounds__(256) void edge_linear_scatter(
    const float* __restrict__ ef, const float* __restrict__ W,
    const float* __restrict__ bias, const int* __restrict__ dst,
    float* __restrict__ msg, int E) {
  int wid  = (int)(((long)blockIdx.x * blockDim.x + threadIdx.x) >> 5);
  int lane = threadIdx.x & 31;
  int rowTiles = (E + 15) >> 4;
  if (wid >= rowTiles * (LATENT / 16)) return;
  int rt = wid >> 2, ct = wid & 3;
  int row0 = rt << 4, col0 = ct << 4;
  int lrow = lane & 15;

  v8f acc;
  float bval = bias[col0 + lrow];
#pragma unroll
  for (int r = 0; r < 8; ++r) acc[r] = bval;

  int arow = min(row0 + lrow, E - 1);
  acc = wmma_f32_gemm<FE_>(ef + ((long)arow - lrow) * FE_, FE_, W + col0, acc, lane);

  int mh = (lane >= 16) ? 8 : 0;
#pragma unroll
  for (int r = 0; r < 8; ++r) {
    int e = row0 + r + mh;
    if (e < E) atomicAdd(&msg[(long)dst[e] * LATENT + col0 + lrow], acc[r]);
  }
}

// ---------------------------------------------------------------------------
// pool[dst] += h[src]   (per-edge gather + scatter-add, float4 granularity)
// ---------------------------------------------------------------------------
__global__ __launch_bounds__(256) void n2n_scatter(
    const float* __restrict__ h, const int* __restrict__ src,
    const int* __restrict__ dst, float* __restrict__ pool, int E) {
  long t = (long)blockIdx.x * blockDim.x + threadIdx.x;
  if (t >= (long)E * 16) return;
  int e  = (int)(t >> 4);
  int c4 = (int)(t & 15) * 4;
  int s = src[e], d = dst[e];
  float4 v = *(const float4*)(h + (long)s * LATENT + c4);
  float* o = pool + (long)d * LATENT + c4;
  atomicAdd(o + 0, v.x);
  atomicAdd(o + 1, v.y);
  atomicAdd(o + 2, v.z);
  atomicAdd(o + 3, v.w);
}

// ---------------------------------------------------------------------------
// h = relu(pool @ W_conv[lv] + b_conv[lv] + msg)
// ---------------------------------------------------------------------------
__global__ __launch_bounds__(256) void conv_wmma(
    const float* __restrict__ pool, const float* __restrict__ Wc,
    const float* __restrict__ bc, const float* __restrict__ msg,
    float* __restrict__ h, int N) {
  int wid  = (int)(((long)blockIdx.x * blockDim.x + threadIdx.x) >> 5);
  int lane = threadIdx.x & 31;
  int rowTiles = (N + 15) >> 4;
  if (wid >= rowTiles * (LATENT / 16)) return;
  int rt = wid >> 2, ct = wid & 3;
  int row0 = rt << 4, col0 = ct << 4;
  int lrow = lane & 15;

  v8f acc;
  float bval = bc[col0 + lrow];
#pragma unroll
  for (int r = 0; r < 8; ++r) acc[r] = bval;

  int arow = min(row0 + lrow, N - 1);
  acc = wmma_f32_gemm<LATENT>(pool + ((long)arow - lrow) * LATENT, LATENT,
                              Wc + col0, acc, lane);

  int mh = (lane >= 16) ? 8 : 0;
#pragma unroll
  for (int r = 0; r < 8; ++r) {
    int m = row0 + r + mh;
    if (m < N) {
      long idx = (long)m * LATENT + col0 + lrow;
      h[idx] = fmaxf(acc[r] + msg[idx], 0.f);
    }
  }
}

// ---------------------------------------------------------------------------
// out = relu(h @ W_fp + b_fp) ; y[graph_id[m]] += out[m]  (fused sum-pool)
// ---------------------------------------------------------------------------
__global__ __launch_bounds__(256) void fp_pool_wmma(
    const float* __restrict__ h, const float* __restrict__ W,
    const float* __restrict__ bias, const int* __restrict__ gid,
    float* __restrict__ y, int N) {
  int wid  = (int)(((long)blockIdx.x * blockDim.x + threadIdx.x) >> 5);
  int lane = threadIdx.x & 31;
  int rowTiles = (N + 15) >> 4;
  if (wid >= rowTiles * (LATENT / 16)) return;
  int rt = wid >> 2, ct = wid & 3;
  int row0 = rt << 4, col0 = ct << 4;
  int lrow = lane & 15;

  v8f acc;
  float bval = bias[col0 + lrow];
#pragma unroll
  for (int r = 0; r < 8; ++r) acc[r] = bval;

  int arow = min(row0 + lrow, N - 1);
  acc = wmma_f32_gemm<LATENT>(h + ((long)arow - lrow) * LATENT, LATENT,
                              W + col0, acc, lane);

  int mh = (lane >= 16) ? 8 : 0;
#pragma unroll
  for (int r = 0; r < 8; ++r) {
    int m = row0 + r + mh;
    if (m < N) {
      float v = fmaxf(acc[r], 0.f);
      atomicAdd(&y[(long)gid[m] * LATENT + col0 + lrow], v);
    }
  }
}

// ---------------------------------------------------------------------------
// Host-side launcher
// ---------------------------------------------------------------------------
extern "C" void kernel_launch(void* const* d_in, const int* in_sizes, int n_in,
                              void* d_out, int out_size, void* d_ws, size_t ws_size,
                              hipStream_t stream) {
  const float* node_feat = (const float*)d_in[0];
  const float* edge_feat = (const float*)d_in[1];
  const int*   edge_src  = (const int*)d_in[2];
  const int*   edge_dst  = (const int*)d_in[3];
  const int*   graph_id  = (const int*)d_in[4];
  const float* W_n2l     = (const float*)d_in[5];
  const float* b_n2l     = (const float*)d_in[6];
  const float* W_e2l     = (const float*)d_in[7];
  const float* b_e2l     = (const float*)d_in[8];
  const float* W_conv    = (const float*)d_in[9];
  const float* b_conv    = (const float*)d_in[10];
  const float* W_fp      = (const float*)d_in[11];
  const float* b_fp      = (const float*)d_in[12];

  const int N = in_sizes[0] / FN_;
  const int E = in_sizes[2];

  float* msg  = (float*)d_ws;                 // [N, 64] input_message
  float* h    = msg  + (size_t)N * LATENT;    // [N, 64] hidden state
  float* pool = h    + (size_t)N * LATENT;    // [N, 64] n2n scatter buffer

  const int TPB = 256;
  long nodeWaves = (long)((N + 15) / 16) * (LATENT / 16);
  long edgeWaves = (long)((E + 15) / 16) * (LATENT / 16);
  int nodeBlocks = (int)((nodeWaves * 32 + TPB - 1) / TPB);
  int edgeBlocks = (int)((edgeWaves * 32 + TPB - 1) / TPB);
  long nd4 = (long)N * LATENT / 4;
  int nd4Blocks = (int)((nd4 + TPB - 1) / TPB);

  // 1) msg = node_feat @ W_n2l + b_n2l
  node_linear_wmma<<<nodeBlocks, TPB, 0, stream>>>(node_feat, W_n2l, b_n2l, msg, N);

  // 2) msg += scatter_dst(edge_feat @ W_e2l + b_e2l)
  edge_linear_scatter<<<edgeBlocks, TPB, 0, stream>>>(edge_feat, W_e2l, b_e2l,
                                                      edge_dst, msg, E);

  // 3) h = relu(msg)
  relu_kernel<<<nd4Blocks, TPB, 0, stream>>>(msg, h, nd4);

  // 4) mean-field iterations
  long sThreads = (long)E * 16;
  int sBlocks = (int)((sThreads + TPB - 1) / TPB);
  for (int lv = 0; lv < LV_; ++lv) {
    zero_kernel<<<nd4Blocks, TPB, 0, stream>>>(pool, nd4);
    n2n_scatter<<<sBlocks, TPB, 0, stream>>>(h, edge_src, edge_dst, pool, E);
    conv_wmma<<<nodeBlocks, TPB, 0, stream>>>(pool,
                                              W_conv + (size_t)lv * LATENT * LATENT,
                                              b_conv + (size_t)lv * LATENT,
                                              msg, h, N);
  }

  // 5) y = segment_sum(relu(h @ W_fp + b_fp), graph_id)
  long g4 = (long)out_size / 4;
  int gBlocks = (int)((g4 + TPB - 1) / TPB);
  zero_kernel<<<gBlocks, TPB, 0, stream>>>((float*)d_out, g4);
  fp_pool_wmma<<<nodeBlocks, TPB, 0, stream>>>(h, W_fp, b_fp, graph_id,
                                               (float*)d_out, N);
}